// SimpleGNN_74517682586571
// MI455X (gfx1250) — compile-verified
//
#include <hip/hip_runtime.h>
#include <hip/hip_bf16.h>

typedef __attribute__((ext_vector_type(2))) float v2f;
typedef __attribute__((ext_vector_type(8))) float v8f;

#define NCLS 10

// ---------------- degree / norm ----------------
__global__ void deg_init_k(float* __restrict__ deg, int n) {
  int i = blockIdx.x * blockDim.x + threadIdx.x;
  if (i < n) deg[i] = 1.0f;  // self loop
}
__global__ void deg_edge_k(const int* __restrict__ col, float* __restrict__ deg, int e) {
  int i = blockIdx.x * blockDim.x + threadIdx.x;
  if (i < e) atomicAdd(&deg[col[i]], 1.0f);
}
__global__ void deg_fin_k(float* __restrict__ deg, int n) {
  int i = blockIdx.x * blockDim.x + threadIdx.x;
  if (i < n) deg[i] = rsqrtf(deg[i]);  // deg >= 1 always
}

// ---------------- GEMM1: h1 = x @ W1^T  (N x 128) x (128 x 128) ----------------
// wave32; each wave computes a 16x128 tile via V_WMMA_F32_16X16X4_F32.
// W1 staged in LDS with XOR swizzle k ^ (4*(o&15)) -> conflict-free float2 reads.
__global__ __launch_bounds__(256) void gemm1_k(const float* __restrict__ x,
                                               const float* __restrict__ W1,
                                               float* __restrict__ h1, int n) {
  __shared__ float Wlds[128 * 128];
  const int tid = threadIdx.x;
  for (int f = tid * 4; f < 128 * 128; f += 256 * 4) {
    int o = f >> 7, k = f & 127;
    float4 v = *(const float4*)(W1 + f);
    int pos = (o << 7) + (k ^ ((o & 15) << 2));  // swizzle is mult-of-4: float4 stays contiguous
    *(float4*)(&Wlds[pos]) = v;
  }
  __syncthreads();

  const int wave = tid >> 5, lane = tid & 31;
  const int tile = blockIdx.x * 8 + wave;
  const int row0 = tile * 16;
  if (row0 >= n) return;  // wave-uniform: EXEC all-ones inside

  const int m  = lane & 15;   // A: M row / B: N col held by this lane
  const int hh = lane >> 4;   // half selects K pair {0,1} vs {2,3}
  const int sw = m << 2;      // B swizzle = 4*((nt*16+m)&15) = 4*m
  const float* xrow = x + (size_t)(row0 + m) * 128;

  v8f acc[8] = {};
  for (int k0 = 0; k0 < 128; k0 += 4) {
    const int ka = k0 + 2 * hh;                 // K = ka, ka+1 for this lane
    float2 av = *(const float2*)(xrow + ka);    // A[m][ka], A[m][ka+1]
    v2f a; a[0] = av.x; a[1] = av.y;
#pragma unroll
    for (int nt = 0; nt < 8; ++nt) {
      // B[k][ncol] = W1[ncol][k], ncol = nt*16 + m
      float2 bv = *(const float2*)(&Wlds[((nt * 16 + m) << 7) + (ka ^ sw)]);
      v2f b; b[0] = bv.x; b[1] = bv.y;
      acc[nt] = __builtin_amdgcn_wmma_f32_16x16x4_f32(
          false, a, false, b, (short)0, acc[nt], false, false);
    }
  }
#pragma unroll
  for (int nt = 0; nt < 8; ++nt) {
#pragma unroll
    for (int i = 0; i < 8; ++i) {
      int r = row0 + i + 8 * hh;  // C/D layout: VGPR i -> rows i and i+8
      if (r < n) h1[(size_t)r * 128 + nt * 16 + m] = acc[nt][i];
    }
  }
}

// ---------------- GEMM2: h2 = o1 @ W2^T, padded to 16 cols ----------------
__global__ __launch_bounds__(256) void gemm2_k(const float* __restrict__ o1,
                                               const float* __restrict__ W2,
                                               float* __restrict__ h2, int n) {
  __shared__ float Wlds[16 * 128];
  const int tid = threadIdx.x;
  for (int f = tid; f < 16 * 128; f += 256) {
    int o = f >> 7, k = f & 127;
    float v = (o < NCLS) ? W2[o * 128 + k] : 0.0f;  // zero-pad rows 10..15
    Wlds[(o << 7) + (k ^ ((o & 15) << 2))] = v;
  }
  __syncthreads();

  const int wave = tid >> 5, lane = tid & 31;
  const int tile = blockIdx.x * 8 + wave;
  const int row0 = tile * 16;
  if (row0 >= n) return;

  const int m  = lane & 15;
  const int hh = lane >> 4;
  const int sw = m << 2;
  const float* xrow = o1 + (size_t)(row0 + m) * 128;

  v8f acc = {};
  for (int k0 = 0; k0 < 128; k0 += 4) {
    const int ka = k0 + 2 * hh;
    float2 av = *(const float2*)(xrow + ka);
    v2f a; a[0] = av.x; a[1] = av.y;
    float2 bv = *(const float2*)(&Wlds[(m << 7) + (ka ^ sw)]);
    v2f b; b[0] = bv.x; b[1] = bv.y;
    acc = __builtin_amdgcn_wmma_f32_16x16x4_f32(
        false, a, false, b, (short)0, acc, false, false);
  }
#pragma unroll
  for (int i = 0; i < 8; ++i) {
    int r = row0 + i + 8 * hh;
    if (r < n) h2[(size_t)r * 16 + m] = acc[i];
  }
}

// ---------------- aggregation layer 1 ----------------
// self-loop + bias: o1 = h1 * dis^2 + b1
__global__ void init1_k(const float* __restrict__ h1, const float* __restrict__ dis,
                        const float* __restrict__ b1, float* __restrict__ o1, int total) {
  int t = blockIdx.x * blockDim.x + threadIdx.x;
  if (t < total) {
    int i = t >> 7, c = t & 127;
    float d = dis[i];
    o1[t] = h1[t] * d * d + b1[c];
  }
}
// one wave per edge: gather 512B row, scatter 128 f32 atomics
__global__ __launch_bounds__(256) void agg1_k(const int* __restrict__ row,
                                              const int* __restrict__ col,
                                              const float* __restrict__ dis,
                                              const float* __restrict__ h1,
                                              float* __restrict__ o1, int e) {
  int w = blockIdx.x * 8 + (threadIdx.x >> 5);
  int lane = threadIdx.x & 31;
  if (w >= e) return;
  int r = row[w], c = col[w];
  float nrm = dis[r] * dis[c];
  float4 v = *(const float4*)(h1 + (size_t)r * 128 + lane * 4);
  float* dst = o1 + (size_t)c * 128 + lane * 4;
  atomicAdd(dst + 0, nrm * v.x);
  atomicAdd(dst + 1, nrm * v.y);
  atomicAdd(dst + 2, nrm * v.z);
  atomicAdd(dst + 3, nrm * v.w);
}

// ---------------- aggregation layer 2 (10 channels) ----------------
__global__ void init2_k(const float* __restrict__ h2, const float* __restrict__ dis,
                        const float* __restrict__ b2, float* __restrict__ out, int n) {
  int t = blockIdx.x * blockDim.x + threadIdx.x;
  if (t < n * 16) {
    int i = t >> 4, c = t & 15;
    if (c < NCLS) {
      float d = dis[i];
      out[i * NCLS + c] = h2[t] * d * d + b2[c];
    }
  }
}
__global__ void agg2_k(const int* __restrict__ row, const int* __restrict__ col,
                       const float* __restrict__ dis, const float* __restrict__ h2,
                       float* __restrict__ out, int e) {
  int t = blockIdx.x * blockDim.x + threadIdx.x;
  int ei = t >> 4, c = t & 15;
  if (ei < e && c < NCLS) {
    int r = row[ei], cn = col[ei];
    float nrm = dis[r] * dis[cn];
    atomicAdd(&out[(size_t)cn * NCLS + c], nrm * h2[(size_t)r * 16 + c]);
  }
}

extern "C" void kernel_launch(void* const* d_in, const int* in_sizes, int n_in,
                              void* d_out, int out_size, void* d_ws, size_t ws_size,
                              hipStream_t stream) {
  const float* x  = (const float*)d_in[0];
  const int*   ei = (const int*)d_in[1];
  const float* W1 = (const float*)d_in[2];
  const float* b1 = (const float*)d_in[3];
  const float* W2 = (const float*)d_in[4];
  const float* b2 = (const float*)d_in[5];
  float* out = (float*)d_out;

  const int n = in_sizes[0] / 128;   // 100000
  const int e = in_sizes[1] / 2;     // 1600000
  const int* row = ei;
  const int* col = ei + e;

  // workspace layout (floats): dis | h1 | o1 | h2  (~109 MB)
  float* ws  = (float*)d_ws;
  float* dis = ws;
  float* h1  = ws + (size_t)((n + 255) & ~255);
  float* o1  = h1 + (size_t)n * 128;
  float* h2  = o1 + (size_t)n * 128;

  deg_init_k<<<(n + 255) / 256, 256, 0, stream>>>(dis, n);
  deg_edge_k<<<(e + 255) / 256, 256, 0, stream>>>(col, dis, e);
  deg_fin_k <<<(n + 255) / 256, 256, 0, stream>>>(dis, n);

  const int tiles   = (n + 15) / 16;
  const int gblocks = (tiles + 7) / 8;

  gemm1_k<<<gblocks, 256, 0, stream>>>(x, W1, h1, n);
  init1_k<<<(n * 128 + 255) / 256, 256, 0, stream>>>(h1, dis, b1, o1, n * 128);
  agg1_k <<<(e + 7) / 8, 256, 0, stream>>>(row, col, dis, h1, o1, e);

  gemm2_k<<<gblocks, 256, 0, stream>>>(o1, W2, h2, n);
  init2_k<<<(n * 16 + 255) / 256, 256, 0, stream>>>(h2, dis, b2, out, n);
  agg2_k <<<(e * 16 + 255) / 256, 256, 0, stream>>>(row, col, dis, h2, out, e);
}